// GNN_Attention_74912819577042
// MI455X (gfx1250) — compile-verified
//
#include <hip/hip_runtime.h>
#include <hip/hip_bf16.h>

typedef float v2f __attribute__((ext_vector_type(2)));
typedef float v8f __attribute__((ext_vector_type(8)));
typedef unsigned int u32x4 __attribute__((ext_vector_type(4)));
typedef int i32x4 __attribute__((ext_vector_type(4)));
typedef int i32x8 __attribute__((ext_vector_type(8)));

#define NEG_SLOPE 0.2f

__device__ __forceinline__ v8f wmma_f32_k4(v2f a, v2f b, v8f c) {
    // D = A(16x4,f32) * B(4x16,f32) + C(16x16,f32)
    return __builtin_amdgcn_wmma_f32_16x16x4_f32(
        /*neg_a=*/false, a, /*neg_b=*/false, b,
        /*c_mod=*/(short)0, c, /*reuse_a=*/false, /*reuse_b=*/false);
}

// order-preserving float -> uint encoding (for atomic max on floats)
__device__ __forceinline__ unsigned enc_f(float f) {
    unsigned b = __float_as_uint(f);
    return (b & 0x80000000u) ? ~b : (b | 0x80000000u);
}
__device__ __forceinline__ float dec_f(unsigned k) {
    unsigned b = (k & 0x80000000u) ? (k & 0x7fffffffu) : ~k;
    return __uint_as_float(b);
}

__device__ __forceinline__ float lrelu(float v) {
    return fmaxf(v, NEG_SLOPE * v);   // valid for 0 < slope < 1
}

// ---------------------------------------------------------------------------
// xl = x @ Wl.T + bl ; xr = x @ Wr.T + br      (N x 8) @ (8 x 128)
// wave-per-16-row tile; two K=4 fp32 WMMAs per 16x16 tile, 8 column tiles.
// FULL specialization removes all per-lane guards (wave-uniform decision).
// ---------------------------------------------------------------------------
template <bool FULL>
__device__ __forceinline__ void proj_tiles(const float* __restrict__ W,
                                           const float* __restrict__ bias,
                                           float* __restrict__ y,
                                           int r0, int m, int k2, int N,
                                           v2f a0, v2f a1) {
    for (int ct = 0; ct < 8; ++ct) {
        const int col = ct * 16 + m;
        const v2f* pw = (const v2f*)(W + (size_t)col * 8);
        v2f b0 = pw[k2];
        v2f b1 = pw[2 + k2];
        v8f acc = {};
        acc = wmma_f32_k4(a0, b0, acc);
        acc = wmma_f32_k4(a1, b1, acc);
        const float bv = bias[col];
#pragma unroll
        for (int r = 0; r < 8; ++r) {
            const int rr = r0 + r + (k2 << 3);
            if (FULL || rr < N) y[(size_t)rr * 128 + col] = acc[r] + bv;
        }
    }
}

__global__ void proj_kernel(const float* __restrict__ x,
                            const float* __restrict__ Wl, const float* __restrict__ bl,
                            const float* __restrict__ Wr, const float* __restrict__ br,
                            float* __restrict__ xl, float* __restrict__ xr, int N) {
    const int wave = (blockIdx.x * blockDim.x + threadIdx.x) >> 5;
    const int lane = threadIdx.x & 31;
    const int r0 = wave * 16;
    if (r0 >= N) return;

    const int m  = lane & 15;       // row within tile (A), col within tile (B/D)
    const int k2 = lane >> 4;       // 0: K pair {0,1}; 1: K pair {2,3}
    const int row = r0 + m;
    const bool full = (r0 + 16 <= N);   // wave-uniform

    v2f a0 = {0.f, 0.f}, a1 = {0.f, 0.f};
    if (full || row < N) {
        const v2f* px = (const v2f*)(x + (size_t)row * 8);
        a0 = px[k2];        // K = {0,1} or {2,3}
        a1 = px[2 + k2];    // K = {4,5} or {6,7}
    }

    if (full) {
        proj_tiles<true >(Wl, bl, xl, r0, m, k2, N, a0, a1);
        proj_tiles<true >(Wr, br, xr, r0, m, k2, N, a0, a1);
    } else {
        proj_tiles<false>(Wl, bl, xl, r0, m, k2, N, a0, a1);
        proj_tiles<false>(Wr, br, xr, r0, m, k2, N, a0, a1);
    }
}

// ---------------------------------------------------------------------------
// per-edge: logits[e][h] = sum_c att[h][c] * lrelu(xl[s] + xr[d] + ea @ We.T)
// + atomic max per (dst, head)
// ---------------------------------------------------------------------------
__global__ void edge_logits_kernel(const int* __restrict__ ei, const float* __restrict__ ea,
                                   const float* __restrict__ xl, const float* __restrict__ xr,
                                   const float* __restrict__ We, const float* __restrict__ att,
                                   float* __restrict__ logits, unsigned* __restrict__ maxenc, int E) {
    const int e = blockIdx.x * blockDim.x + threadIdx.x;
    if (e >= E) return;
    const int s = ei[e];
    const int d = ei[(size_t)E + e];
    const float ea0 = ea[(size_t)e * 2];
    const float ea1 = ea[(size_t)e * 2 + 1];

    const float4* pl = (const float4*)(xl + (size_t)s * 128);
    const float4* pr = (const float4*)(xr + (size_t)d * 128);
    const float4* pa = (const float4*)att;   // att flattened (H*C = 128)
    const float4* pw = (const float4*)We;    // We is (128, 2) row-major

    float l0 = 0.f, l1 = 0.f;
#pragma unroll 4
    for (int t = 0; t < 16; ++t) {           // head 0: channels 0..63
        float4 L = pl[t], R = pr[t], A = pa[t];
        float4 W0 = pw[2 * t], W1 = pw[2 * t + 1];
        float v0 = lrelu(L.x + R.x + ea0 * W0.x + ea1 * W0.y);
        float v1 = lrelu(L.y + R.y + ea0 * W0.z + ea1 * W0.w);
        float v2 = lrelu(L.z + R.z + ea0 * W1.x + ea1 * W1.y);
        float v3 = lrelu(L.w + R.w + ea0 * W1.z + ea1 * W1.w);
        l0 += A.x * v0 + A.y * v1 + A.z * v2 + A.w * v3;
    }
#pragma unroll 4
    for (int t = 16; t < 32; ++t) {          // head 1: channels 64..127
        float4 L = pl[t], R = pr[t], A = pa[t];
        float4 W0 = pw[2 * t], W1 = pw[2 * t + 1];
        float v0 = lrelu(L.x + R.x + ea0 * W0.x + ea1 * W0.y);
        float v1 = lrelu(L.y + R.y + ea0 * W0.z + ea1 * W0.w);
        float v2 = lrelu(L.z + R.z + ea0 * W1.x + ea1 * W1.y);
        float v3 = lrelu(L.w + R.w + ea0 * W1.z + ea1 * W1.w);
        l1 += A.x * v0 + A.y * v1 + A.z * v2 + A.w * v3;
    }
    logits[(size_t)e * 2]     = l0;
    logits[(size_t)e * 2 + 1] = l1;
    atomicMax(maxenc + (size_t)d * 2,     enc_f(l0));
    atomicMax(maxenc + (size_t)d * 2 + 1, enc_f(l1));
}

// ---------------------------------------------------------------------------
// p = exp(logit - max[dst]); stash p in alpha output region; accumulate denom
// ---------------------------------------------------------------------------
__global__ void edge_exp_kernel(const int* __restrict__ ei, const float* __restrict__ logits,
                                const unsigned* __restrict__ maxenc,
                                float* __restrict__ denom, float* __restrict__ alpha, int E) {
    const int e = blockIdx.x * blockDim.x + threadIdx.x;
    if (e >= E) return;
    const int d = ei[(size_t)E + e];
    const float m0 = dec_f(maxenc[(size_t)d * 2]);
    const float m1 = dec_f(maxenc[(size_t)d * 2 + 1]);
    const float p0 = __expf(logits[(size_t)e * 2]     - m0);
    const float p1 = __expf(logits[(size_t)e * 2 + 1] - m1);
    alpha[(size_t)e * 2]     = p0;
    alpha[(size_t)e * 2 + 1] = p1;
    atomicAdd(denom + (size_t)d * 2,     p0);
    atomicAdd(denom + (size_t)d * 2 + 1, p1);
}

// ---------------------------------------------------------------------------
// wave-per-edge: finalize alpha, edge weight ew + degree, message scatter
// ---------------------------------------------------------------------------
__global__ void edge_msg_kernel(const int* __restrict__ ei, const float* __restrict__ xl,
                                const float* __restrict__ denom, float* __restrict__ alpha,
                                float* __restrict__ ew, float* __restrict__ deg,
                                float* __restrict__ h, int E) {
    const int gid  = blockIdx.x * blockDim.x + threadIdx.x;
    const int e    = gid >> 5;
    const int lane = gid & 31;
    if (e >= E) return;
    const int s = ei[e];
    const int d = ei[(size_t)E + e];
    const float a0 = alpha[(size_t)e * 2]     / denom[(size_t)d * 2];
    const float a1 = alpha[(size_t)e * 2 + 1] / denom[(size_t)d * 2 + 1];
    if (lane == 0) {
        alpha[(size_t)e * 2]     = a0;
        alpha[(size_t)e * 2 + 1] = a1;
        const float w = 0.5f * (a0 + a1);
        ew[e] = w;
        atomicAdd(deg + d, w);
    }
    const float4 v = ((const float4*)(xl + (size_t)s * 128))[lane];
    const float av = (lane < 16) ? a0 : a1;
    float* hd = h + (size_t)d * 128 + lane * 4;
    atomicAdd(hd + 0, v.x * av);
    atomicAdd(hd + 1, v.y * av);
    atomicAdd(hd + 2, v.z * av);
    atomicAdd(hd + 3, v.w * av);
}

// ---------------------------------------------------------------------------
// h = relu(h + gat_bias); dinv = deg > 0 ? rsqrt(deg) : 0     (32 lanes/node)
// ---------------------------------------------------------------------------
__global__ void node_mid_kernel(float* __restrict__ h, const float* __restrict__ gat_bias,
                                const float* __restrict__ deg, float* __restrict__ dinv, int N) {
    const int gid = blockIdx.x * blockDim.x + threadIdx.x;
    const int n = gid >> 5, q = gid & 31;
    if (n >= N) return;
    float4* ph = (float4*)(h + (size_t)n * 128);
    const float4 b = ((const float4*)gat_bias)[q];
    float4 v = ph[q];
    v.x = fmaxf(v.x + b.x, 0.f);
    v.y = fmaxf(v.y + b.y, 0.f);
    v.z = fmaxf(v.z + b.z, 0.f);
    v.w = fmaxf(v.w + b.w, 0.f);
    ph[q] = v;
    if (q == 0) {
        const float dg = deg[n];
        dinv[n] = dg > 0.f ? rsqrtf(dg) : 0.f;
    }
}

// ---------------------------------------------------------------------------
// hw = h @ gcn_W.T        (N x 128) @ (128 x 64), fp32 WMMA, K = 32 x 4
// gcn_W staged into LDS once per workgroup by the Tensor Data Mover; B
// fragments then come from LDS (ds_load) instead of per-wave L2 traffic.
// ---------------------------------------------------------------------------
template <bool FULL>
__device__ __forceinline__ void gemm_hw_body(const float* __restrict__ h,
                                             const float* ldsW,
                                             float* __restrict__ hw,
                                             int r0, int m, int k2, int N) {
    const int row = r0 + m;
    const v2f zero2 = {0.f, 0.f};
    const v2f* ha = (const v2f*)(h + (size_t)row * 128);
    const bool rv = FULL || (row < N);

    for (int ct = 0; ct < 4; ++ct) {
        const int col = ct * 16 + m;
        const v2f* wb = (const v2f*)(ldsW + (size_t)col * 128);
        v8f acc = {};
#pragma unroll
        for (int kq = 0; kq < 32; ++kq) {    // K block of 4 per step
            v2f a = FULL ? ha[kq * 2 + k2] : (rv ? ha[kq * 2 + k2] : zero2);
            v2f b = wb[kq * 2 + k2];
            acc = wmma_f32_k4(a, b, acc);
        }
#pragma unroll
        for (int r = 0; r < 8; ++r) {
            const int rr = r0 + r + (k2 << 3);
            if (FULL || rr < N) hw[(size_t)rr * 64 + col] = acc[r];
        }
    }
}

__global__ void gemm_hw_kernel(const float* __restrict__ h, const float* __restrict__ W,
                               float* __restrict__ hw, int N) {
    __shared__ float ldsW[64 * 128];   // 32 KB: full gcn_W (64 rows x 128)

    // --- stage gcn_W into LDS with the Tensor Data Mover (one op per WG) ---
    if ((threadIdx.x >> 5) == 0) {     // wave 0 only (TDM ignores EXEC)
        const unsigned lds_off = (unsigned)(uintptr_t)(&ldsW[0]);
        const unsigned long long ga = (unsigned long long)(uintptr_t)W;
        // D# group 0: count=1 | lds_addr | global_addr | type=2 ("image")
        u32x4 g0 = { 1u,
                     lds_off,
                     (unsigned)ga,
                     (unsigned)(ga >> 32) | (2u << 30) };
        // D# group 1: data_size=4B, tensor 128x64, tile 128x64, dim0 stride 128
        i32x8 g1 = { (int)0x00020000u,        // w0: data_size=2 (4 bytes)
                     (int)(128u << 16),       // w1: tensor_dim0[15:0]=128
                     (int)(64u  << 16),       // w2: tensor_dim1[15:0]=64
                     (int)(128u << 16),       // w3: tile_dim0=128
                     (int)64,                 // w4: tile_dim1=64, tile_dim2=0
                     (int)128,                // w5: tensor_dim0_stride=128
                     0, 0 };                  // w6/w7: high stride bits = 0
        i32x4 gz4 = { 0, 0, 0, 0 };           // groups 2/3: unused (2-D tensor)
        i32x8 gz8 = { 0, 0, 0, 0, 0, 0, 0, 0 };
        __builtin_amdgcn_tensor_load_to_lds(g0, g1, gz4, gz4, gz8, 0);
        __builtin_amdgcn_s_wait_tensorcnt(0);
    }
    __syncthreads();                   // publish LDS tile to all 8 waves

    const int wave = (blockIdx.x * blockDim.x + threadIdx.x) >> 5;
    const int lane = threadIdx.x & 31;
    const int r0 = wave * 16;
    if (r0 >= N) return;
    const int m  = lane & 15;
    const int k2 = lane >> 4;
    if (r0 + 16 <= N) gemm_hw_body<true >(h, ldsW, hw, r0, m, k2, N);
    else              gemm_hw_body<false>(h, ldsW, hw, r0, m, k2, N);
}

// ---------------------------------------------------------------------------
// wave-per-edge: h2[dst] += hw[src] * (dinv[src] * ew * dinv[dst])
// ---------------------------------------------------------------------------
__global__ void gcn_edge_kernel(const int* __restrict__ ei, const float* __restrict__ hw,
                                const float* __restrict__ ew, const float* __restrict__ dinv,
                                float* __restrict__ h2, int E) {
    const int gid  = blockIdx.x * blockDim.x + threadIdx.x;
    const int e    = gid >> 5;
    const int lane = gid & 31;
    if (e >= E) return;
    const int s = ei[e];
    const int d = ei[(size_t)E + e];
    const float norm = dinv[s] * ew[e] * dinv[d];
    const float2 v = ((const float2*)(hw + (size_t)s * 64))[lane];
    float* pd = h2 + (size_t)d * 64 + lane * 2;
    atomicAdd(pd,     v.x * norm);
    atomicAdd(pd + 1, v.y * norm);
}

// ---------------------------------------------------------------------------
// out = relu(h2 + gcn_b) @ out_W.T + out_b
// ---------------------------------------------------------------------------
__global__ void out_kernel(const float* __restrict__ h2, const float* __restrict__ gcb,
                           const float* __restrict__ oW, const float* __restrict__ ob,
                           float* __restrict__ out, int N) {
    const int n = blockIdx.x * blockDim.x + threadIdx.x;
    if (n >= N) return;
    const float4* p  = (const float4*)(h2 + (size_t)n * 64);
    const float4* pb = (const float4*)gcb;
    const float4* w0 = (const float4*)oW;
    const float4* w1 = (const float4*)(oW + 64);
    float acc0 = 0.f, acc1 = 0.f;
#pragma unroll
    for (int t = 0; t < 16; ++t) {
        float4 v = p[t], b = pb[t], a = w0[t], c = w1[t];
        float r0 = fmaxf(v.x + b.x, 0.f);
        float r1 = fmaxf(v.y + b.y, 0.f);
        float r2 = fmaxf(v.z + b.z, 0.f);
        float r3 = fmaxf(v.w + b.w, 0.f);
        acc0 += a.x * r0 + a.y * r1 + a.z * r2 + a.w * r3;
        acc1 += c.x * r0 + c.y * r1 + c.z * r2 + c.w * r3;
    }
    out[(size_t)n * 2]     = acc0 + ob[0];
    out[(size_t)n * 2 + 1] = acc1 + ob[1];
}

// ---------------------------------------------------------------------------
extern "C" void kernel_launch(void* const* d_in, const int* in_sizes, int n_in,
                              void* d_out, int out_size, void* d_ws, size_t ws_size,
                              hipStream_t stream) {
    const float* x   = (const float*)d_in[0];
    const float* ea  = (const float*)d_in[1];
    const int*   ei  = (const int*)d_in[2];
    const float* Wl  = (const float*)d_in[3];
    const float* bl  = (const float*)d_in[4];
    const float* Wr  = (const float*)d_in[5];
    const float* br  = (const float*)d_in[6];
    const float* We  = (const float*)d_in[7];
    const float* att = (const float*)d_in[8];
    const float* gb  = (const float*)d_in[9];
    const float* gW  = (const float*)d_in[10];
    const float* gcb = (const float*)d_in[11];
    const float* oW  = (const float*)d_in[12];
    const float* ob  = (const float*)d_in[13];

    const int N = in_sizes[0] / 8;   // IN = 8
    const int E = in_sizes[1] / 2;   // ED = 2

    float* ws = (float*)d_ws;
    size_t off = 0;
    auto take = [&](size_t nel) {
        float* p = ws + off;
        off += (nel + 3) & ~(size_t)3;   // keep 16B alignment for float4
        return p;
    };
    float*    xl     = take((size_t)N * 128);
    float*    xr     = take((size_t)N * 128);
    float*    hwv    = take((size_t)N * 64);
    float*    logits = take((size_t)E * 2);
    float*    ewv    = take((size_t)E);
    float*    dinv   = take((size_t)N);
    const size_t zstart = off;            // zero-initialized region below
    float*    hbuf   = take((size_t)N * 128);
    float*    h2     = take((size_t)N * 64);
    float*    denom  = take((size_t)N * 2);
    float*    deg    = take((size_t)N);
    unsigned* maxenc = (unsigned*)take((size_t)N * 2);  // 0 == encoded "-inf"
    const size_t zbytes = (off - zstart) * sizeof(float);

    (void)hipMemsetAsync(ws + zstart, 0, zbytes, stream);

    float* out   = (float*)d_out;
    float* alpha = out + (size_t)N * 2;   // outputs: (out[N,2], alpha[E,2]) flat

    const int waves  = (N + 15) / 16;
    const int pblks  = (waves + 7) / 8;   // 8 waves (256 thr) per block

    proj_kernel<<<pblks, 256, 0, stream>>>(x, Wl, bl, Wr, br, xl, xr, N);
    edge_logits_kernel<<<(E + 255) / 256, 256, 0, stream>>>(ei, ea, xl, xr, We, att,
                                                            logits, maxenc, E);
    edge_exp_kernel<<<(E + 255) / 256, 256, 0, stream>>>(ei, logits, maxenc, denom, alpha, E);
    edge_msg_kernel<<<(E + 7) / 8, 256, 0, stream>>>(ei, xl, denom, alpha, ewv, deg, hbuf, E);
    node_mid_kernel<<<(int)(((size_t)N * 32 + 255) / 256), 256, 0, stream>>>(hbuf, gb, deg, dinv, N);
    gemm_hw_kernel<<<pblks, 256, 0, stream>>>(hbuf, gW, hwv, N);
    gcn_edge_kernel<<<(E + 7) / 8, 256, 0, stream>>>(ei, hwv, ewv, dinv, h2, E);
    out_kernel<<<(N + 255) / 256, 256, 0, stream>>>(h2, gcb, oW, ob, out, N);
}